// RWKV_RNN_13030930776672
// MI455X (gfx1250) — compile-verified
//
#include <hip/hip_runtime.h>
#include <math.h>

typedef float v2f __attribute__((ext_vector_type(2)));
typedef float v4f __attribute__((ext_vector_type(4)));
typedef float v8f __attribute__((ext_vector_type(8)));

#define D_  1024
#define H_  16
#define S_  64
#define F_  3584
#define V_  65536
#define L_  8
#define SCALE_ 0.125f
#define MV_WAVES 8

// ---------------------------------------------------------------- post-ops
__device__ __forceinline__ float postop(float y, int mode) {
    if (mode == 1) return y / (1.f + __expf(-y));                 // silu
    if (mode == 2) { float t = y > 0.f ? y : 0.f; return t * t; } // relu^2
    if (mode == 3) return 1.f / (1.f + __expf(-y));               // sigmoid
    return y;
}

// ---------------------------------------------------------------- layernorm
__device__ __forceinline__ void ln_body(const float* __restrict__ xin,
                                        const float* __restrict__ w,
                                        const float* __restrict__ b,
                                        float* __restrict__ out,
                                        float* red) {
    const int tid = threadIdx.x;
    float s = 0.f;
    for (int i = tid; i < D_; i += 256) s += xin[i];
    red[tid] = s; __syncthreads();
    for (int off = 128; off > 0; off >>= 1) {
        if (tid < off) red[tid] += red[tid + off];
        __syncthreads();
    }
    const float mean = red[0] * (1.f / D_);
    __syncthreads();
    float vs = 0.f;
    for (int i = tid; i < D_; i += 256) { float t = xin[i] - mean; vs += t * t; }
    red[tid] = vs; __syncthreads();
    for (int off = 128; off > 0; off >>= 1) {
        if (tid < off) red[tid] += red[tid + off];
        __syncthreads();
    }
    const float rstd = rsqrtf(red[0] * (1.f / D_) + 1e-5f);
    __syncthreads();
    for (int i = tid; i < D_; i += 256)
        out[i] = (xin[i] - mean) * rstd * w[i] + b[i];
}

__global__ __launch_bounds__(256) void ln_kernel(const float* __restrict__ xin,
                                                 const float* __restrict__ w,
                                                 const float* __restrict__ b,
                                                 float* __restrict__ out) {
    __shared__ float red[256];
    ln_body(xin, w, b, out, red);
}

__global__ __launch_bounds__(256) void embed_ln_kernel(const float* __restrict__ emb,
                                                       const int* __restrict__ token,
                                                       const float* __restrict__ w,
                                                       const float* __restrict__ b,
                                                       float* __restrict__ out) {
    __shared__ float red[256];
    const float* xin = emb + (long)token[0] * D_;
    ln_body(xin, w, b, out, red);
}

// ---------------------------------------------------------------- token/ffn mixes
__global__ void mix4_kernel(const float* __restrict__ xn, const float* __restrict__ s,
                            const float* __restrict__ tmk, const float* __restrict__ tmv,
                            const float* __restrict__ tmr, const float* __restrict__ tmg,
                            float* __restrict__ xk, float* __restrict__ xv,
                            float* __restrict__ xr, float* __restrict__ xg) {
    const int i = blockIdx.x * blockDim.x + threadIdx.x;
    const float a = xn[i], st = s[i];
    float t;
    t = tmk[i]; xk[i] = a * t + st * (1.f - t);
    t = tmv[i]; xv[i] = a * t + st * (1.f - t);
    t = tmr[i]; xr[i] = a * t + st * (1.f - t);
    t = tmg[i]; xg[i] = a * t + st * (1.f - t);
}

__global__ void mix2_kernel(const float* __restrict__ xn, const float* __restrict__ s,
                            const float* __restrict__ tmk, const float* __restrict__ tmr,
                            float* __restrict__ xk, float* __restrict__ xr) {
    const int i = blockIdx.x * blockDim.x + threadIdx.x;
    const float a = xn[i], st = s[i];
    float t;
    t = tmk[i]; xk[i] = a * t + st * (1.f - t);
    t = tmr[i]; xr[i] = a * t + st * (1.f - t);
}

// ---------------------------------------------------------------- WMMA matvec
// Up to 4 independent matvecs per launch (gridDim.y selects the slice).
struct MVArgs {
    const float* W;
    const float* xin;
    float*       y;
    int nk;      // K length (multiple of 128)
    int nrows;   // rows (multiple of 16)
    int mode;    // post-op
    int accum;   // y += result instead of y =
};
struct MV4 { MVArgs s[4]; };

// y[16*bx + m] = postop( sum_k W[row,k] * x[k] )  via V_WMMA_F32_16X16X4_F32.
// K-slot permutation: WMMA0 slots = {k,k+1 | k+4,k+5}, WMMA1 = {k+2,k+3 | k+6,k+7},
// so each lane's A data for 2 WMMAs is ONE contiguous global b128 at
// wrow + k + 4*(lane>=16), and the matching B (x broadcast to all 16 cols,
// identical within each lane-half) is ONE ds b128 at the same K offset.
// 4 accumulators / 16 K per iteration break the WMMA RAW chain.
__global__ __launch_bounds__(256) void matvec_wmma(MV4 p) {
    const MVArgs a = p.s[blockIdx.y];
    if ((int)blockIdx.x * 16 >= a.nrows) return;

    __shared__ float lx[F_];            // up to 3584 inputs (14 KB of 320 KB LDS)
    __shared__ float lpart[MV_WAVES * 16];
    const int tid = threadIdx.x;
    const int nk  = a.nk;
    for (int i = tid; i < nk; i += 256) lx[i] = a.xin[i];
    __syncthreads();

    const int wave = tid >> 5;
    const int lane = tid & 31;
    const int hib  = lane >> 4;             // 0: lanes 0-15, 1: lanes 16-31
    const int m    = lane & 15;
    const long rowBase = (long)blockIdx.x * 16;
    const float* wrow = a.W + (rowBase + m) * (long)nk + 4 * hib;
    const float* lxo  = lx + 4 * hib;

    const int kc = nk / MV_WAVES;           // 128 or 448 (multiple of 16)
    const int k0 = wave * kc;

    v8f acc0 = {}, acc1 = {}, acc2 = {}, acc3 = {};
    for (int k = k0; k < k0 + kc; k += 16) {
        v4f w0 = *(const v4f*)(wrow + k);
        v4f w1 = *(const v4f*)(wrow + k + 8);
        v4f x0 = *(const v4f*)(lxo + k);
        v4f x1 = *(const v4f*)(lxo + k + 8);
        v2f aa, bb;
        aa = __builtin_shufflevector(w0, w0, 0, 1);
        bb = __builtin_shufflevector(x0, x0, 0, 1);
        acc0 = __builtin_amdgcn_wmma_f32_16x16x4_f32(false, aa, false, bb,
                                                     (short)0, acc0, false, false);
        aa = __builtin_shufflevector(w0, w0, 2, 3);
        bb = __builtin_shufflevector(x0, x0, 2, 3);
        acc1 = __builtin_amdgcn_wmma_f32_16x16x4_f32(false, aa, false, bb,
                                                     (short)0, acc1, false, false);
        aa = __builtin_shufflevector(w1, w1, 0, 1);
        bb = __builtin_shufflevector(x1, x1, 0, 1);
        acc2 = __builtin_amdgcn_wmma_f32_16x16x4_f32(false, aa, false, bb,
                                                     (short)0, acc2, false, false);
        aa = __builtin_shufflevector(w1, w1, 2, 3);
        bb = __builtin_shufflevector(x1, x1, 2, 3);
        acc3 = __builtin_amdgcn_wmma_f32_16x16x4_f32(false, aa, false, bb,
                                                     (short)0, acc3, false, false);
    }
    v8f acc = (acc0 + acc1) + (acc2 + acc3);

    if (m == 0) {                            // lane 0 -> rows 0..7, lane 16 -> rows 8..15
        const int base = wave * 16 + hib * 8;
#pragma unroll
        for (int j = 0; j < 8; ++j) lpart[base + j] = acc[j];
    }
    __syncthreads();

    if (tid < 16) {
        float y = 0.f;
#pragma unroll
        for (int w2 = 0; w2 < MV_WAVES; ++w2) y += lpart[w2 * 16 + tid];
        y = postop(y, a.mode);
        const long row = rowBase + tid;
        if (a.accum) a.y[row] += y; else a.y[row] = y;
    }
}

// ---------------------------------------------------------------- wkv (per head)
// wkv[h,n] = sum_s r[h,s] * ( tf[h,s]*k[h,s]*v[h,n]*SCALE + st[h,s,n]*SCALE )
__global__ void wkv_kernel(const float* __restrict__ k, const float* __restrict__ v,
                           const float* __restrict__ r, const float* __restrict__ tf,
                           const float* __restrict__ st, float* __restrict__ wkv) {
    const int h = blockIdx.x, n = threadIdx.x;
    const float* kh  = k  + h * S_;
    const float* rh  = r  + h * S_;
    const float* tfh = tf + h * S_;
    const float* sh  = st + (long)h * S_ * S_;
    const float vn = v[h * S_ + n] * SCALE_;
    float acc = 0.f;
#pragma unroll 4
    for (int s = 0; s < S_; ++s)
        acc += rh[s] * (tfh[s] * kh[s] * vn + sh[s * S_ + n] * SCALE_);
    wkv[h * S_ + n] = acc;
}

// ---------------------------------------------------------------- groupnorm * gate
__global__ void gn_gate_kernel(const float* __restrict__ wkv, const float* __restrict__ w,
                               const float* __restrict__ b, const float* __restrict__ g,
                               float* __restrict__ out) {
    __shared__ float red[S_];
    const int h = blockIdx.x, n = threadIdx.x;
    const float v0 = wkv[h * S_ + n];
    red[n] = v0; __syncthreads();
    for (int off = 32; off > 0; off >>= 1) {
        if (n < off) red[n] += red[n + off];
        __syncthreads();
    }
    const float mean = red[0] * (1.f / S_);
    __syncthreads();
    const float t = v0 - mean;
    red[n] = t * t; __syncthreads();
    for (int off = 32; off > 0; off >>= 1) {
        if (n < off) red[n] += red[n + off];
        __syncthreads();
    }
    const float rstd = rsqrtf(red[0] * (1.f / S_) + 1e-5f);
    const int i = h * S_ + n;
    out[i] = (t * rstd * w[i] + b[i]) * g[i];
}

// ---------------------------------------------------------------- ffn combine
__global__ void fuse_kernel(float* __restrict__ x, const float* __restrict__ fr,
                            const float* __restrict__ fv) {
    const int i = blockIdx.x * blockDim.x + threadIdx.x;
    x[i] += fr[i] * fv[i];
}

// ---------------------------------------------------------------- host helpers
static inline MVArgs mv(const float* W, const float* xin, float* y,
                        int nk, int nrows, int mode, int accum) {
    MVArgs a; a.W = W; a.xin = xin; a.y = y;
    a.nk = nk; a.nrows = nrows; a.mode = mode; a.accum = accum;
    return a;
}

// ---------------------------------------------------------------- host driver
extern "C" void kernel_launch(void* const* d_in, const int* in_sizes, int n_in,
                              void* d_out, int out_size, void* d_ws, size_t ws_size,
                              hipStream_t stream) {
    const int*   token   = (const int*)  d_in[0];
    const float* emb_w   = (const float*)d_in[1];
    const float* ln0_w   = (const float*)d_in[2];
    const float* ln0_b   = (const float*)d_in[3];
    const float* ln1_w   = (const float*)d_in[4];
    const float* ln1_b   = (const float*)d_in[5];
    const float* ln2_w   = (const float*)d_in[6];
    const float* ln2_b   = (const float*)d_in[7];
    const float* tmk     = (const float*)d_in[8];
    const float* tmv     = (const float*)d_in[9];
    const float* tmr     = (const float*)d_in[10];
    const float* tmg     = (const float*)d_in[11];
    /* d_in[12] time_decay_raw: dead (new states are discarded) */
    const float* tf      = (const float*)d_in[13];
    const float* att_kw  = (const float*)d_in[14];
    const float* att_vw  = (const float*)d_in[15];
    const float* att_rw  = (const float*)d_in[16];
    const float* att_gw  = (const float*)d_in[17];
    const float* att_ow  = (const float*)d_in[18];
    const float* ln_x_w  = (const float*)d_in[19];
    const float* ln_x_b  = (const float*)d_in[20];
    const float* ftmk    = (const float*)d_in[21];
    const float* ftmr    = (const float*)d_in[22];
    const float* ffn_kw  = (const float*)d_in[23];
    const float* ffn_vw  = (const float*)d_in[24];
    const float* ffn_rw  = (const float*)d_in[25];
    const float* lnout_w = (const float*)d_in[26];
    const float* lnout_b = (const float*)d_in[27];
    const float* head_w  = (const float*)d_in[28];
    const float* s_att   = (const float*)d_in[29];
    const float* s_wkv   = (const float*)d_in[30];
    const float* s_ffn   = (const float*)d_in[31];
    float* out = (float*)d_out;

    float* ws  = (float*)d_ws;
    float* x    = ws + 0;
    float* xn   = ws + 1024;
    float* xk   = ws + 2048;
    float* xv   = ws + 3072;
    float* xr   = ws + 4096;
    float* xg   = ws + 5120;
    float* kk   = ws + 6144;
    float* vv   = ws + 7168;
    float* rr   = ws + 8192;
    float* gg   = ws + 9216;
    float* wk   = ws + 10240;
    float* aa   = ws + 11264;
    float* fxk  = ws + 12288;
    float* fxr  = ws + 13312;
    float* fk   = ws + 14336;   // 3584
    float* fv   = ws + 17920;
    float* fr   = ws + 18944;

    embed_ln_kernel<<<1, 256, 0, stream>>>(emb_w, token, ln0_w, ln0_b, x);

    for (int l = 0; l < L_; ++l) {
        const long dOff = (long)l * D_;
        const long wOff = (long)l * D_ * D_;
        // ---- time mixing ----
        ln_kernel<<<1, 256, 0, stream>>>(x, ln1_w + dOff, ln1_b + dOff, xn);
        mix4_kernel<<<4, 256, 0, stream>>>(xn, s_att + dOff, tmk + dOff, tmv + dOff,
                                           tmr + dOff, tmg + dOff, xk, xv, xr, xg);
        {   // k,v,r,g matvecs fused: 256 blocks in flight
            MV4 p;
            p.s[0] = mv(att_kw + wOff, xk, kk, D_, D_, 0, 0);
            p.s[1] = mv(att_vw + wOff, xv, vv, D_, D_, 0, 0);
            p.s[2] = mv(att_rw + wOff, xr, rr, D_, D_, 0, 0);
            p.s[3] = mv(att_gw + wOff, xg, gg, D_, D_, 1, 0);   // silu
            matvec_wmma<<<dim3(D_ / 16, 4), 256, 0, stream>>>(p);
        }
        wkv_kernel<<<H_, S_, 0, stream>>>(kk, vv, rr, tf + (long)l * H_ * S_,
                                          s_wkv + (long)l * H_ * S_ * S_, wk);
        gn_gate_kernel<<<H_, S_, 0, stream>>>(wk, ln_x_w + dOff, ln_x_b + dOff, gg, aa);
        {   // x += ow @ a
            MV4 p; p.s[0] = mv(att_ow + wOff, aa, x, D_, D_, 0, 1);
            p.s[1] = p.s[2] = p.s[3] = p.s[0];
            matvec_wmma<<<dim3(D_ / 16, 1), 256, 0, stream>>>(p);
        }
        // ---- channel mixing ----
        ln_kernel<<<1, 256, 0, stream>>>(x, ln2_w + dOff, ln2_b + dOff, xn);
        mix2_kernel<<<4, 256, 0, stream>>>(xn, s_ffn + dOff, ftmk + dOff, ftmr + dOff,
                                           fxk, fxr);
        {   // fr = sigmoid(frw@fxr)  ||  fk = relu(fkw@fxk)^2, concurrently
            MV4 p;
            p.s[0] = mv(ffn_kw + (long)l * F_ * D_, fxk, fk, D_, F_, 2, 0);
            p.s[1] = mv(ffn_rw + wOff,              fxr, fr, D_, D_, 3, 0);
            p.s[2] = p.s[3] = p.s[1];
            matvec_wmma<<<dim3(F_ / 16, 2), 256, 0, stream>>>(p);
        }
        {   // fv = fvw @ fk
            MV4 p; p.s[0] = mv(ffn_vw + (long)l * D_ * F_, fk, fv, F_, D_, 0, 0);
            p.s[1] = p.s[2] = p.s[3] = p.s[0];
            matvec_wmma<<<dim3(D_ / 16, 1), 256, 0, stream>>>(p);
        }
        fuse_kernel<<<4, 256, 0, stream>>>(x, fr, fv);
    }

    ln_kernel<<<1, 256, 0, stream>>>(x, lnout_w, lnout_b, xn);
    {   // logits = head_w @ ln(x)
        MV4 p; p.s[0] = mv(head_w, xn, out, D_, V_, 0, 0);
        p.s[1] = p.s[2] = p.s[3] = p.s[0];
        matvec_wmma<<<dim3(V_ / 16, 1), 256, 0, stream>>>(p);
    }
}